// CircuitGNN_DualTask_12429635355176
// MI455X (gfx1250) — compile-verified
//
#include <hip/hip_runtime.h>
#include <math.h>

#define N_NODES 20000
#define N_EDGES 320000
#define ET_EDGES (N_EDGES + N_NODES)   // with self loops
#define SLOPE 0.2f
#define EPS_BN 1e-5f

typedef float v2f __attribute__((ext_vector_type(2)));
typedef float v8f __attribute__((ext_vector_type(8)));

// ---------------------------------------------------------------------------
// fp32 WMMA GEMM: C[M x NCOLS] = A[M x K] @ B[K x NCOLS] + bias (row broadcast)
// Compile-time K, NCOLS -> fully unrolled K-loop, no guards, EXEC all ones.
// grid = (M/16, NCOLS/64), block = (32, 4): each wave owns one 16x16 tile.
// A 16x4 frag (ISA 7.12.2): lanes 0-15 hold K=k,k+1; lanes 16-31 hold K=k+2,k+3
//   -> per lane one aligned 8B (v2f) load.
// B 4x16 frag: VGPR0 rows K=k (lanes 0-15) / K=k+2 (lanes 16-31), VGPR1 K=k+1/k+3.
// C/D: VGPR j -> row (j + 8*(lane>=16)), col = lane&15.
// M = 20000 = 1250*16 exactly -> no row bounds checks.
// ---------------------------------------------------------------------------
template<int K, int NCOLS>
__global__ __launch_bounds__(128) void gemm_wmma_f32(
    const float* __restrict__ A, const float* __restrict__ B,
    const float* __restrict__ bias, float* __restrict__ C)
{
  const int lane    = threadIdx.x;                       // 0..31 (wave32)
  const int rowBase = blockIdx.x * 16;
  const int colBase = (blockIdx.y * 4 + threadIdx.y) * 16;
  const int mn      = lane & 15;                         // row for A, col for B
  const int khalf   = (lane >> 4) << 1;                  // 0 or 2

  const float* arow = A + (rowBase + mn) * K + khalf;    // 8B-aligned (K%4==0, khalf even)
  const float* bcol = B + colBase + mn;

  v8f c = {};
#pragma unroll
  for (int k = 0; k < K; k += 4) {
    const v2f a = *(const v2f*)(arow + k);
    v2f b;
    b.x = bcol[(k + khalf) * NCOLS];
    b.y = bcol[(k + khalf + 1) * NCOLS];
    c = __builtin_amdgcn_wmma_f32_16x16x4_f32(false, a, false, b,
                                              (short)0, c, false, false);
  }
  const float bv     = bias[colBase + mn];
  const int   rowOff = (lane >> 4) * 8;
#pragma unroll
  for (int j = 0; j < 8; ++j)
    C[(rowBase + rowOff + j) * NCOLS + colBase + mn] = c[j] + bv;
}

// ---------------------------------------------------------------------------
// layer-0 zero padding: x[N,17] -> xpad[N,20]; W[17,128] -> Wpad[20,128]
// ---------------------------------------------------------------------------
__global__ void pad_x(const float* __restrict__ src, float* __restrict__ dst)
{
  const int i = blockIdx.x * blockDim.x + threadIdx.x;
  if (i >= N_NODES * 20) return;
  const int r = i / 20, c = i - r * 20;
  dst[i] = (c < 17) ? src[r * 17 + c] : 0.0f;
}

__global__ void pad_w(const float* __restrict__ src, float* __restrict__ dst)
{
  const int i = blockIdx.x * blockDim.x + threadIdx.x;
  if (i >= 20 * 128) return;
  dst[i] = (i < 17 * 128) ? src[i] : 0.0f;
}

// ---------------------------------------------------------------------------
// segment-softmax helpers
// ---------------------------------------------------------------------------
__device__ inline void atomicMaxF(float* addr, float val) {
  unsigned int* ua  = (unsigned int*)addr;
  unsigned int  old = *ua;
  while (__uint_as_float(old) < val) {
    unsigned int assumed = old;
    old = atomicCAS(ua, assumed, __float_as_uint(val));
    if (old == assumed) break;
  }
}

__global__ void init_seg(float* __restrict__ mx, float* __restrict__ sm,
                         float* __restrict__ acc, int accCount)
{
  const int i = blockIdx.x * blockDim.x + threadIdx.x;
  if (i < N_NODES * 2) { mx[i] = -3.402823466e38f; sm[i] = 0.0f; }
  if (i < accCount)    { acc[i] = 0.0f; }
}

// one wave per edge; lanes stride channels; H=2 hardcoded
__global__ void edge_logit(const float* __restrict__ xl, const float* __restrict__ xr,
                           const float* __restrict__ att,
                           const int* __restrict__ src, const int* __restrict__ dst,
                           float* __restrict__ logit, float* __restrict__ mx, int C)
{
  const int e = blockIdx.x * 8 + (threadIdx.x >> 5);
  if (e >= ET_EDGES) return;
  const int lane = threadIdx.x & 31;
  const int s = (e < N_EDGES) ? src[e] : (e - N_EDGES);
  const int d = (e < N_EDGES) ? dst[e] : (e - N_EDGES);
  const int HC = 2 * C;
  float a0 = 0.0f, a1 = 0.0f;
  for (int cc = lane; cc < C; cc += 32) {
    float v0 = xl[s * HC + cc] + xr[d * HC + cc];
    v0 = (v0 > 0.0f) ? v0 : SLOPE * v0;
    a0 += v0 * att[cc];
    float v1 = xl[s * HC + C + cc] + xr[d * HC + C + cc];
    v1 = (v1 > 0.0f) ? v1 : SLOPE * v1;
    a1 += v1 * att[C + cc];
  }
#pragma unroll
  for (int off = 16; off > 0; off >>= 1) {
    a0 += __shfl_xor(a0, off, 32);
    a1 += __shfl_xor(a1, off, 32);
  }
  if (lane == 0) {
    logit[e * 2 + 0] = a0;
    logit[e * 2 + 1] = a1;
    atomicMaxF(&mx[d * 2 + 0], a0);
    atomicMaxF(&mx[d * 2 + 1], a1);
  }
}

__global__ void edge_expsum(const float* __restrict__ logit, const float* __restrict__ mx,
                            const int* __restrict__ dst, float* __restrict__ sm)
{
  const int t = blockIdx.x * blockDim.x + threadIdx.x;
  if (t >= ET_EDGES * 2) return;
  const int e = t >> 1, hh = t & 1;
  const int d = (e < N_EDGES) ? dst[e] : (e - N_EDGES);
  atomicAdd(&sm[d * 2 + hh], expf(logit[t] - mx[d * 2 + hh]));
}

__global__ void edge_accum(const float* __restrict__ xl, const float* __restrict__ logit,
                           const float* __restrict__ mx, const float* __restrict__ sm,
                           const int* __restrict__ src, const int* __restrict__ dst,
                           float* __restrict__ acc, int C)
{
  const int e = blockIdx.x * 8 + (threadIdx.x >> 5);
  if (e >= ET_EDGES) return;
  const int lane = threadIdx.x & 31;
  const int s = (e < N_EDGES) ? src[e] : (e - N_EDGES);
  const int d = (e < N_EDGES) ? dst[e] : (e - N_EDGES);
  const int HC = 2 * C;
  const float al0 = expf(logit[e * 2 + 0] - mx[d * 2 + 0]) / sm[d * 2 + 0];
  const float al1 = expf(logit[e * 2 + 1] - mx[d * 2 + 1]) / sm[d * 2 + 1];
  for (int cc = lane; cc < C; cc += 32) {
    atomicAdd(&acc[d * HC + cc],     al0 * xl[s * HC + cc]);
    atomicAdd(&acc[d * HC + C + cc], al1 * xl[s * HC + C + cc]);
  }
}

// out[n][c] = mean over heads of acc[n][h][c] + bias[c]
__global__ void head_mean(const float* __restrict__ acc, const float* __restrict__ bias,
                          float* __restrict__ out, int C)
{
  const int i = blockIdx.x * blockDim.x + threadIdx.x;
  if (i >= N_NODES * C) return;
  const int node = i / C, c = i - node * C;
  out[i] = 0.5f * (acc[node * 2 * C + c] + acc[node * 2 * C + C + c]) + bias[c];
}

// per-column batch stats (training-mode BN, biased variance)
__global__ void col_stats(const float* __restrict__ X, int C,
                          float* __restrict__ mean, float* __restrict__ var)
{
  __shared__ float ssum[256], ssq[256];
  const int c = blockIdx.x;
  float s = 0.0f, q = 0.0f;
  for (int r = threadIdx.x; r < N_NODES; r += blockDim.x) {
    const float v = X[r * C + c];
    s += v; q += v * v;
  }
  ssum[threadIdx.x] = s; ssq[threadIdx.x] = q;
  __syncthreads();
  for (int stride = 128; stride > 0; stride >>= 1) {
    if (threadIdx.x < stride) {
      ssum[threadIdx.x] += ssum[threadIdx.x + stride];
      ssq[threadIdx.x]  += ssq[threadIdx.x + stride];
    }
    __syncthreads();
  }
  if (threadIdx.x == 0) {
    const float mu = ssum[0] / (float)N_NODES;
    mean[c] = mu;
    var[c]  = ssq[0] / (float)N_NODES - mu * mu;
  }
}

__global__ void bn_relu(const float* __restrict__ X,
                        const float* __restrict__ mean, const float* __restrict__ var,
                        const float* __restrict__ g, const float* __restrict__ b,
                        const float* __restrict__ resid, float* __restrict__ out, int C)
{
  const int i = blockIdx.x * blockDim.x + threadIdx.x;
  if (i >= N_NODES * C) return;
  const int c = i % C;
  float v = g[c] * (X[i] - mean[c]) * rsqrtf(var[c] + EPS_BN) + b[c];
  v = (v > 0.0f) ? v : 0.0f;
  out[i] = resid ? (resid[i] + v) : v;
}

__global__ void final_heads(const float* __restrict__ h2,
                            const float* __restrict__ Wimp, const float* __restrict__ bimp,
                            const float* __restrict__ Wpol, const float* __restrict__ bpol,
                            float* __restrict__ out)
{
  const int i = blockIdx.x * blockDim.x + threadIdx.x;
  if (i >= N_NODES) return;
  float a = 0.0f, p = 0.0f;
#pragma unroll
  for (int c = 0; c < 32; ++c) {
    const float v = h2[i * 32 + c];
    a += v * Wimp[c];
    p += v * Wpol[c];
  }
  out[i]           = a + bimp[0];
  out[N_NODES + i] = 1.0f / (1.0f + expf(-(p + bpol[0])));
}

// ---------------------------------------------------------------------------
// host-side layer driver (all launches on `stream`, graph-capture safe)
// ---------------------------------------------------------------------------
template<int K, int C>
static void run_gat_layer(const float* xin,
                          const float* Wl, const float* bl,
                          const float* Wr, const float* br,
                          const float* att, const float* bias,
                          const float* g, const float* be,
                          const float* resid, float* hout,
                          float* xl, float* xr, float* logit, float* mx,
                          float* sm, float* acc, float* outb, float* stats,
                          const int* src, const int* dst, hipStream_t stream)
{
  constexpr int HC = 2 * C;
  dim3 gblock(32, 4);
  dim3 ggrid(N_NODES / 16, HC / 64);
  gemm_wmma_f32<K, HC><<<ggrid, gblock, 0, stream>>>(xin, Wl, bl, xl);
  gemm_wmma_f32<K, HC><<<ggrid, gblock, 0, stream>>>(xin, Wr, br, xr);

  const int accCount = N_NODES * HC;
  init_seg<<<(accCount + 255) / 256, 256, 0, stream>>>(mx, sm, acc, accCount);

  const int eblocks = (ET_EDGES + 7) / 8;
  edge_logit<<<eblocks, 256, 0, stream>>>(xl, xr, att, src, dst, logit, mx, C);
  edge_expsum<<<(ET_EDGES * 2 + 255) / 256, 256, 0, stream>>>(logit, mx, dst, sm);
  edge_accum<<<eblocks, 256, 0, stream>>>(xl, logit, mx, sm, src, dst, acc, C);

  head_mean<<<(N_NODES * C + 255) / 256, 256, 0, stream>>>(acc, bias, outb, C);
  col_stats<<<C, 256, 0, stream>>>(outb, C, stats, stats + 64);
  bn_relu<<<(N_NODES * C + 255) / 256, 256, 0, stream>>>(outb, stats, stats + 64,
                                                         g, be, resid, hout, C);
}

extern "C" void kernel_launch(void* const* d_in, const int* in_sizes, int n_in,
                              void* d_out, int out_size, void* d_ws, size_t ws_size,
                              hipStream_t stream) {
  const float* x    = (const float*)d_in[0];
  const int*   src  = (const int*)d_in[1];
  const int*   dst  = (const int*)d_in[2];
  const float* W0l  = (const float*)d_in[3];
  const float* b0l  = (const float*)d_in[4];
  const float* W0r  = (const float*)d_in[5];
  const float* b0r  = (const float*)d_in[6];
  const float* att0 = (const float*)d_in[7];
  const float* bias0= (const float*)d_in[8];
  const float* g0   = (const float*)d_in[9];
  const float* be0  = (const float*)d_in[10];
  const float* Wml  = (const float*)d_in[11];
  const float* bml  = (const float*)d_in[12];
  const float* Wmr  = (const float*)d_in[13];
  const float* bmr  = (const float*)d_in[14];
  const float* attm = (const float*)d_in[15];
  const float* biasm= (const float*)d_in[16];
  const float* gm   = (const float*)d_in[17];
  const float* bem  = (const float*)d_in[18];
  const float* WLl  = (const float*)d_in[19];
  const float* bLl  = (const float*)d_in[20];
  const float* WLr  = (const float*)d_in[21];
  const float* bLr  = (const float*)d_in[22];
  const float* attL = (const float*)d_in[23];
  const float* biasL= (const float*)d_in[24];
  const float* gL   = (const float*)d_in[25];
  const float* beL  = (const float*)d_in[26];
  const float* Wimp = (const float*)d_in[27];
  const float* bimp = (const float*)d_in[28];
  const float* Wpol = (const float*)d_in[29];
  const float* bpol = (const float*)d_in[30];

  // workspace carve-up (floats)
  float* ws    = (float*)d_ws;
  float* hbuf  = ws;                         // N*64
  float* xl    = hbuf  + N_NODES * 64;       // N*128
  float* xr    = xl    + N_NODES * 128;      // N*128
  float* logit = xr    + N_NODES * 128;      // ET*2
  float* mx    = logit + ET_EDGES * 2;       // N*2
  float* sm    = mx    + N_NODES * 2;        // N*2
  float* acc   = sm    + N_NODES * 2;        // N*128
  float* outb  = acc   + N_NODES * 128;      // N*64
  float* stats = outb  + N_NODES * 64;       // 128 (mean|var)
  float* h2    = stats + 128;                // N*32
  float* xpad  = h2    + N_NODES * 32;       // N*20
  float* W0lp  = xpad  + N_NODES * 20;       // 20*128
  float* W0rp  = W0lp  + 20 * 128;           // 20*128

  // layer-0 zero padding (K: 17 -> 20)
  pad_x<<<(N_NODES * 20 + 255) / 256, 256, 0, stream>>>(x, xpad);
  pad_w<<<(20 * 128 + 255) / 256, 256, 0, stream>>>(W0l, W0lp);
  pad_w<<<(20 * 128 + 255) / 256, 256, 0, stream>>>(W0r, W0rp);

  // layer 0: xpad[N,20] -> h[N,64], no residual
  run_gat_layer<20, 64>(xpad, W0lp, b0l, W0rp, b0r, att0, bias0, g0, be0,
                        nullptr, hbuf, xl, xr, logit, mx, sm, acc, outb, stats,
                        src, dst, stream);

  // 18 mid layers with residual: h = h + relu(bn(gat(h)))
  for (int i = 0; i < 18; ++i) {
    run_gat_layer<64, 64>(hbuf,
                          Wml + i * 64 * 128, bml + i * 128,
                          Wmr + i * 64 * 128, bmr + i * 128,
                          attm + i * 128, biasm + i * 64, gm + i * 64, bem + i * 64,
                          hbuf, hbuf, xl, xr, logit, mx, sm, acc, outb, stats,
                          src, dst, stream);
  }

  // last layer: h[N,64] -> h2[N,32], no residual
  run_gat_layer<64, 32>(hbuf, WLl, bLl, WLr, bLr, attL, biasL, gL, beL,
                        nullptr, h2, xl, xr, logit, mx, sm, acc, outb, stats,
                        src, dst, stream);

  // output heads
  final_heads<<<(N_NODES + 255) / 256, 256, 0, stream>>>(
      h2, Wimp, bimp, Wpol, bpol, (float*)d_out);
}